// STRankLossStable_24429773979783
// MI455X (gfx1250) — compile-verified
//
#include <hip/hip_runtime.h>
#include <math.h>

// Problem constants (from the reference).
#define DCOLS   8
#define NGROUPS 32768
#define GD      (NGROUPS * DCOLS)

typedef float v4f __attribute__((ext_vector_type(4)));
typedef int   v4i __attribute__((ext_vector_type(4)));

// Order-preserving float<->int map so segment-max is a native
// non-returning global_atomic_max_i32 (no CAS loop).
__device__ __forceinline__ int f2ord(float f) {
    int i = __float_as_int(f);
    return (i >= 0) ? i : (i ^ 0x7fffffff);
}
__device__ __forceinline__ float ord2f(int i) {
    return __int_as_float((i >= 0) ? i : (i ^ 0x7fffffff));
}
#define ORD_NEG_INF 0x807fffff   // f2ord(-inf)

// wave32-aware block reduction -> one global_atomic_add_f64 per block.
__device__ __forceinline__ void block_reduce_add_f64(double v, double* acc) {
    #pragma unroll
    for (int o = 16; o > 0; o >>= 1) v += __shfl_down(v, o, 32);
    __shared__ double wsum[32];               // up to 1024 threads / 32 lanes
    const int lane = threadIdx.x & 31;
    const int wid  = threadIdx.x >> 5;
    if (lane == 0) wsum[wid] = v;
    __syncthreads();
    if (threadIdx.x == 0) {
        double s = 0.0;
        const int nw = (blockDim.x + 31) >> 5;
        for (int i = 0; i < nw; ++i) s += wsum[i];
        atomicAdd(acc, s);                    // global_atomic_add_f64
    }
}

// ---------------------------------------------------------------- pass A: init
__global__ void k_init(int* __restrict__ max_ord, float* __restrict__ sumexp,
                       float* __restrict__ cntsum, double* __restrict__ acc) {
    const int j = blockIdx.x * blockDim.x + threadIdx.x;
    if (j < GD) {
        max_ord[j] = ORD_NEG_INF;
        sumexp[j]  = 0.0f;
        cntsum[j]  = 0.0f;
    }
    if (j < 2) acc[j] = 0.0;
}

// ------------------------------------------------ pass B: per-group column max
// Default (temporal) loads: 144 MB of pred+groups fill the 192 MB L2 and are
// re-read from L2 in pass C.
__global__ void k_segmax(const float* __restrict__ pred,
                         const int* __restrict__ groups,
                         int* __restrict__ max_ord, int nrows) {
    const int row = blockIdx.x * blockDim.x + threadIdx.x;
    if (row >= nrows) return;
    const int g    = groups[row];
    const int base = g * DCOLS;
    __builtin_prefetch(&max_ord[base], 1, 0);           // global_prefetch_b8
    const v4f* p4 = reinterpret_cast<const v4f*>(pred) + (size_t)row * 2;
    const v4f a = p4[0];
    const v4f b = p4[1];
    float p[DCOLS] = {a.x, a.y, a.z, a.w, b.x, b.y, b.z, b.w};
    #pragma unroll
    for (int d = 0; d < DCOLS; ++d)
        atomicMax(&max_ord[base + d], f2ord(p[d]));     // global_atomic_max_i32
}

// ------------------- pass C: sumexp per group, cntsum per group, Σ count·pred
// pred is last-use and count is single-use -> non-temporal loads (TH_NT) so
// the streams don't thrash L2; max_ord (1 MB) stays cache-hot.
__global__ void k_sumexp(const float* __restrict__ pred,
                         const float* __restrict__ count,
                         const int* __restrict__ groups,
                         const int* __restrict__ max_ord,
                         float* __restrict__ sumexp,
                         float* __restrict__ cntsum,
                         double* __restrict__ acc, int nrows) {
    const int row = blockIdx.x * blockDim.x + threadIdx.x;
    double local = 0.0;
    if (row < nrows) {
        const int g    = groups[row];
        const int base = g * DCOLS;
        __builtin_prefetch(&max_ord[base], 0, 0);       // global_prefetch_b8
        const v4f* p4 = reinterpret_cast<const v4f*>(pred)  + (size_t)row * 2;
        const v4f* c4 = reinterpret_cast<const v4f*>(count) + (size_t)row * 2;
        const v4f a  = __builtin_nontemporal_load(p4);
        const v4f b  = __builtin_nontemporal_load(p4 + 1);
        const v4f ca = __builtin_nontemporal_load(c4);
        const v4f cb = __builtin_nontemporal_load(c4 + 1);
        const v4i* m4 = reinterpret_cast<const v4i*>(max_ord + base);
        const v4i m0 = m4[0];
        const v4i m1 = m4[1];
        float p[DCOLS] = {a.x, a.y, a.z, a.w, b.x, b.y, b.z, b.w};
        float c[DCOLS] = {ca.x, ca.y, ca.z, ca.w, cb.x, cb.y, cb.z, cb.w};
        int   m[DCOLS] = {m0.x, m0.y, m0.z, m0.w, m1.x, m1.y, m1.z, m1.w};
        float cp = 0.0f;
        #pragma unroll
        for (int d = 0; d < DCOLS; ++d) {
            const float e = __expf(p[d] - ord2f(m[d])); // v_exp_f32 path
            atomicAdd(&sumexp[base + d], e);            // global_atomic_add_f32
            atomicAdd(&cntsum[base + d], c[d]);         // global_atomic_add_f32
            cp = fmaf(p[d], c[d], cp);
        }
        local = (double)cp;
    }
    block_reduce_add_f64(local, &acc[0]);
}

// --------------- pass D: Σ_{g,d} (log(sumexp)+max)·cntsum  (tiny: 256K elems)
__global__ void k_groupsum(const int* __restrict__ max_ord,
                           const float* __restrict__ sumexp,
                           const float* __restrict__ cntsum,
                           double* __restrict__ acc) {
    const int j = blockIdx.x * blockDim.x + threadIdx.x;
    double local = 0.0;
    if (j < GD) {
        const float s = sumexp[j];
        if (s > 0.0f) {  // skip empty groups (reference never indexes them)
            const float m = ord2f(max_ord[j]);
            local = (double)(logf(s) + m) * (double)cntsum[j];
        }
    }
    block_reduce_add_f64(local, &acc[1]);
}

// ------------------------------------------------------------- final scalar
__global__ void k_finish(const double* __restrict__ acc,
                         float* __restrict__ out, long long total_elems) {
    if (threadIdx.x == 0 && blockIdx.x == 0)
        out[0] = (float)((acc[1] - acc[0]) / (double)total_elems);
}

extern "C" void kernel_launch(void* const* d_in, const int* in_sizes, int n_in,
                              void* d_out, int out_size, void* d_ws, size_t ws_size,
                              hipStream_t stream) {
    const float* pred   = (const float*)d_in[0];
    const float* count  = (const float*)d_in[1];
    const int*   groups = (const int*)d_in[2];
    float*       out    = (float*)d_out;

    const int nrows = in_sizes[2];                 // N (groups is length-N)

    // Workspace layout (needs 3 MB + 16 B; ws base is allocation-aligned).
    char* ws = (char*)d_ws;
    int*    max_ord = (int*)(ws);                          // GD ints   (1 MB)
    float*  sumexp  = (float*)(ws + (size_t)GD * 4);       // GD floats (1 MB)
    float*  cntsum  = (float*)(ws + (size_t)GD * 8);       // GD floats (1 MB)
    double* acc     = (double*)(ws + (size_t)GD * 12);     // 2 doubles

    const int B = 256;                             // 8 wave32 waves per block
    const int gInit = (GD + B - 1) / B;
    const int gRows = (nrows + B - 1) / B;

    k_init    <<<gInit, B, 0, stream>>>(max_ord, sumexp, cntsum, acc);
    k_segmax  <<<gRows, B, 0, stream>>>(pred, groups, max_ord, nrows);
    k_sumexp  <<<gRows, B, 0, stream>>>(pred, count, groups, max_ord,
                                        sumexp, cntsum, acc, nrows);
    k_groupsum<<<gInit, B, 0, stream>>>(max_ord, sumexp, cntsum, acc);
    k_finish  <<<1, 32, 0, stream>>>(acc, out, (long long)nrows * DCOLS);
}